// DigitGlia_2765958938739
// MI455X (gfx1250) — compile-verified
//
#include <hip/hip_runtime.h>
#include <hip/hip_bf16.h>

typedef __attribute__((ext_vector_type(2))) float v2f;
typedef __attribute__((ext_vector_type(8))) float v8f;

// ---------------------------------------------------------------------------
// Hardware tanh (gfx1250 V_TANH_F32). Builtin if the toolchain declares it,
// otherwise inline asm (v_nop appended to cover the TRANS-op result hazard).
// ---------------------------------------------------------------------------
__device__ __forceinline__ float dev_tanh(float x) {
#if __has_builtin(__builtin_amdgcn_tanhf)
    return __builtin_amdgcn_tanhf(x);
#elif __has_builtin(__builtin_amdgcn_tanh_f32)
    return __builtin_amdgcn_tanh_f32(x);
#else
    float r;
    asm volatile("v_tanh_f32 %0, %1\n\tv_nop" : "=v"(r) : "v"(x));
    return r;
#endif
}

// ---------------------------------------------------------------------------
// Fused conv1 + pool + relu + conv2 + pool + relu. One block per image,
// 320 threads. Everything staged in LDS (~30 KB / workgroup).
// Output: flattened [B, 320] with feature index f = c*16 + h*4 + w (NCHW).
// ---------------------------------------------------------------------------
__global__ __launch_bounds__(320) void conv_fused_kernel(
    const float* __restrict__ x,   // [B,1,28,28]
    const float* __restrict__ w1,  // [10,1,5,5]
    const float* __restrict__ b1,  // [10]
    const float* __restrict__ w2,  // [20,10,5,5]
    const float* __restrict__ b2,  // [20]
    float* __restrict__ out)       // [B,320]
{
    __shared__ float s_img[784];    // 28x28
    __shared__ float s_p1[1440];    // [10][12][12] pooled+relu conv1
    __shared__ float s_w1[250];
    __shared__ float s_w2[5000];

    const int b   = blockIdx.x;
    const int tid = threadIdx.x;

    for (int i = tid; i < 784;  i += 320) s_img[i] = x[(size_t)b * 784 + i];
    for (int i = tid; i < 250;  i += 320) s_w1[i] = w1[i];
    for (int i = tid; i < 5000; i += 320) s_w2[i] = w2[i];
    __syncthreads();

    // conv1 (5x5 valid) -> +bias -> maxpool2 -> relu, 10*12*12 = 1440 outputs
    for (int i = tid; i < 1440; i += 320) {
        const int c   = i / 144;
        const int rem = i % 144;
        const int ph  = rem / 12;
        const int pw  = rem % 12;
        const float* wc = &s_w1[c * 25];
        float m = -3.4e38f;
#pragma unroll
        for (int dy = 0; dy < 2; ++dy) {
#pragma unroll
            for (int dx = 0; dx < 2; ++dx) {
                const int oy = 2 * ph + dy;
                const int ox = 2 * pw + dx;
                float acc = 0.f;
#pragma unroll
                for (int ky = 0; ky < 5; ++ky)
#pragma unroll
                    for (int kx = 0; kx < 5; ++kx)
                        acc = fmaf(s_img[(oy + ky) * 28 + (ox + kx)], wc[ky * 5 + kx], acc);
                m = fmaxf(m, acc);
            }
        }
        s_p1[i] = fmaxf(m + b1[c], 0.f);
    }
    __syncthreads();

    // conv2 (5x5 valid over 10 ch) -> +bias -> maxpool2 -> relu; 320 outputs
    const int c  = tid >> 4;
    const int hw = tid & 15;
    const int h  = hw >> 2;
    const int w  = hw & 3;
    const float* w2c = &s_w2[c * 250];
    float m = -3.4e38f;
#pragma unroll
    for (int dy = 0; dy < 2; ++dy) {
#pragma unroll
        for (int dx = 0; dx < 2; ++dx) {
            const int oy = 2 * h + dy;
            const int ox = 2 * w + dx;
            float acc = 0.f;
            for (int ci = 0; ci < 10; ++ci) {
                const float* p  = &s_p1[ci * 144 + oy * 12 + ox];
                const float* wk = &w2c[ci * 25];
#pragma unroll
                for (int ky = 0; ky < 5; ++ky)
#pragma unroll
                    for (int kx = 0; kx < 5; ++kx)
                        acc = fmaf(p[ky * 12 + kx], wk[ky * 5 + kx], acc);
            }
            m = fmaxf(m, acc);
        }
    }
    out[(size_t)b * 320 + tid] = fmaxf(m + b2[c], 0.f);
}

// ---------------------------------------------------------------------------
// Zero a small pad region (>= max K floats) used to make FC loads branchless.
// ---------------------------------------------------------------------------
__global__ void zero_pad_kernel(float* __restrict__ p)
{
    p[threadIdx.x] = 0.f;
}

// ---------------------------------------------------------------------------
// FC layer: out[B x N] = act( in[B x K] @ W[N x K]^T ),  B = 8192, K even.
// fp32 WMMA (v_wmma_f32_16x16x4_f32). Each wave: 64x32 output tile
// (4 M-subtiles x 2 N-subtiles, 8 v8f accumulators), K stepped by 4.
//
// Branchless hot loop: out-of-range N columns stream from a zeroed pad
// buffer (pointer select hoisted out of the loop); the K%4==2 remainder is
// handled by a uniform one-shot tail whose OOB lanes also read the pad.
// Double-buffered loads overlap global_load latency with the WMMA block.
//
// A fragment (16x4 f32, 2 VGPR): lane L: M = L&15, K-pair = (L>>4)*2.
// B fragment (4x16 f32, 2 VGPR): lane L: N = L&15, K-pair = (L>>4)*2.
// D (16x16 f32, 8 VGPR):         lane L: N = L&15, rows (L>>4)*8 .. +7.
// ---------------------------------------------------------------------------
#define FC_WMMA_8(ACC, A, Bf)                                                  \
    _Pragma("unroll")                                                          \
    for (int im = 0; im < 4; ++im)                                             \
        _Pragma("unroll")                                                      \
        for (int jn = 0; jn < 2; ++jn)                                         \
            ACC[im][jn] = __builtin_amdgcn_wmma_f32_16x16x4_f32(               \
                false, A[im], false, Bf[jn], (short)0, ACC[im][jn], false, false);

template <bool APPLY_TANH>
__global__ __launch_bounds__(256) void fc_wmma_kernel(
    const float* __restrict__ in, const float* __restrict__ W,
    const float* __restrict__ zpad,
    float* __restrict__ out, int K, int N)
{
    const int lane = threadIdx.x & 31;
    const int wave = blockIdx.x * 8 + (threadIdx.x >> 5);
    const int tileN = wave >> 7;           // 128 M-tiles (8192/64)
    const int tileM = wave & 127;
    const int row0  = tileM * 64;
    const int col0  = tileN * 32;
    const int mr    = lane & 15;
    const int half  = lane >> 4;

    // Per-lane streaming base pointers, pre-offset by this lane-half's K pair.
    const float* ap[4];
#pragma unroll
    for (int im = 0; im < 4; ++im)
        ap[im] = in + (size_t)(row0 + im * 16 + mr) * K + 2 * half;
    const float* bp[2];
#pragma unroll
    for (int jn = 0; jn < 2; ++jn) {
        const int col = col0 + jn * 16 + mr;
        bp[jn] = (col < N) ? (W + (size_t)col * K + 2 * half) : (zpad + 2 * half);
    }

    v8f acc[4][2];
#pragma unroll
    for (int im = 0; im < 4; ++im)
#pragma unroll
        for (int jn = 0; jn < 2; ++jn)
            acc[im][jn] = (v8f){0.f, 0.f, 0.f, 0.f, 0.f, 0.f, 0.f, 0.f};

    const int Kmain = K & ~3;              // K >= 12 always, so Kmain >= 4

    // Software pipeline: preload chunk 0, then load k while computing k-4.
    v2f a_cur[4], b_cur[2];
#pragma unroll
    for (int im = 0; im < 4; ++im) a_cur[im] = *(const v2f*)(ap[im]);
#pragma unroll
    for (int jn = 0; jn < 2; ++jn) b_cur[jn] = *(const v2f*)(bp[jn]);

    for (int k0 = 4; k0 < Kmain; k0 += 4) {
        v2f a_nxt[4], b_nxt[2];
#pragma unroll
        for (int im = 0; im < 4; ++im) a_nxt[im] = *(const v2f*)(ap[im] + k0);
#pragma unroll
        for (int jn = 0; jn < 2; ++jn) b_nxt[jn] = *(const v2f*)(bp[jn] + k0);

        FC_WMMA_8(acc, a_cur, b_cur);

#pragma unroll
        for (int im = 0; im < 4; ++im) a_cur[im] = a_nxt[im];
#pragma unroll
        for (int jn = 0; jn < 2; ++jn) b_cur[jn] = b_nxt[jn];
    }
    FC_WMMA_8(acc, a_cur, b_cur);

    if (K & 2) {                            // uniform tail: K % 4 == 2
        // half==1 lanes would read K-pair {Kmain+2, Kmain+3} (OOB) -> pad.
        v2f at[4], bt[2];
#pragma unroll
        for (int im = 0; im < 4; ++im) {
            const float* p = half ? zpad
                                  : (in + (size_t)(row0 + im * 16 + mr) * K + Kmain);
            at[im] = *(const v2f*)p;
        }
#pragma unroll
        for (int jn = 0; jn < 2; ++jn) {
            const int col = col0 + jn * 16 + mr;
            const float* p = (!half && col < N) ? (W + (size_t)col * K + Kmain)
                                                : zpad;
            bt[jn] = *(const v2f*)p;
        }
        FC_WMMA_8(acc, at, bt);
    }

    // Epilogue: optional tanh, masked store (store guard is fine to diverge).
#pragma unroll
    for (int im = 0; im < 4; ++im) {
#pragma unroll
        for (int jn = 0; jn < 2; ++jn) {
            const int col = col0 + jn * 16 + mr;
            if (col < N) {
#pragma unroll
                for (int r = 0; r < 8; ++r) {
                    const int row = row0 + im * 16 + half * 8 + r;
                    float v = acc[im][jn][r];
                    if (APPLY_TANH) v = dev_tanh(v);
                    out[(size_t)row * N + col] = v;
                }
            }
        }
    }
}

// ---------------------------------------------------------------------------
// log_softmax over [8192, 10]
// ---------------------------------------------------------------------------
__global__ void log_softmax10_kernel(const float* __restrict__ in,
                                     float* __restrict__ out)
{
    const int b = blockIdx.x * blockDim.x + threadIdx.x;
    if (b >= 8192) return;
    float v[10];
    float m = -3.4e38f;
#pragma unroll
    for (int i = 0; i < 10; ++i) { v[i] = in[b * 10 + i]; m = fmaxf(m, v[i]); }
    float s = 0.f;
#pragma unroll
    for (int i = 0; i < 10; ++i) s += __expf(v[i] - m);
    const float lse = m + __logf(s);
#pragma unroll
    for (int i = 0; i < 10; ++i) out[b * 10 + i] = v[i] - lse;
}

// ---------------------------------------------------------------------------
extern "C" void kernel_launch(void* const* d_in, const int* in_sizes, int n_in,
                              void* d_out, int out_size, void* d_ws, size_t ws_size,
                              hipStream_t stream)
{
    const float* x  = (const float*)d_in[0];
    const float* w1 = (const float*)d_in[1];
    const float* b1 = (const float*)d_in[2];
    const float* w2 = (const float*)d_in[3];
    const float* b2 = (const float*)d_in[4];

    // Workspace layout: two 10.5 MB activation buffers + 2 KB zero pad.
    float* bufA = (float*)d_ws;
    float* bufB = (float*)((char*)d_ws + ((size_t)16 << 20));
    float* zpad = (float*)((char*)d_ws + ((size_t)28 << 20));

    zero_pad_kernel<<<1, 512, 0, stream>>>(zpad);
    conv_fused_kernel<<<8192, 320, 0, stream>>>(x, w1, b1, w2, b2, bufA);

    // Layer output sizes: 318..50 step -2 (135), then 48..10 step -2 (20).
    int sizes[160]; int nl = 0;
    for (int s = 318; s >= 50; s -= 2) sizes[nl++] = s;
    for (int s = 48;  s >= 10; s -= 2) sizes[nl++] = s;

    for (int i = 0; i < nl; ++i) {
        const int N = sizes[i];
        const int Kd = N + 2;                 // chain: 320->318->...->10
        const float* Wp = (const float*)d_in[5 + i];
        const float* src = (i & 1) ? bufB : bufA;
        float*       dst = (i & 1) ? bufA : bufB;
        const int tilesN = (N + 31) / 32;
        dim3 grid(16 * tilesN);               // 8 waves/block, 128 M-tiles
        if (i == nl - 1)
            fc_wmma_kernel<false><<<grid, 256, 0, stream>>>(src, Wp, zpad, dst, Kd, N);
        else
            fc_wmma_kernel<true><<<grid, 256, 0, stream>>>(src, Wp, zpad, dst, Kd, N);
    }

    const float* logits = (nl & 1) ? bufB : bufA;
    log_softmax10_kernel<<<8192 / 256, 256, 0, stream>>>(logits, (float*)d_out);
}